// TE_55044300865691
// MI455X (gfx1250) — compile-verified
//
#include <hip/hip_runtime.h>
#include <stdint.h>

// Problem constants (match reference)
#define SN   128
#define SPK  512
#define RR   8
#define PP   2
#define HH   128
#define WW   128
#define NCOORD (PP*HH*WW)      // 32768
#define TAU_INV 0.01f
#define PFRAC 0.3f
#define GMAXV 1.0f
#define GMINV 0.1f
#define EPS_SCALE 1e-12f

typedef __attribute__((ext_vector_type(4))) unsigned int v4u;
typedef __attribute__((ext_vector_type(8))) int          v8i;
typedef __attribute__((ext_vector_type(4))) int          v4i;

// ---------------------------------------------------------------------------
// Kernel 1: build sparse decay chains per sequence (nodes live in LDS),
// then flatten each of the 8 slot states into (coord, weight) entry lists.
// Node = int4 { parent, coord, bitcast(add), bitcast(m) }, interleaved
// as nodes[i*32 + lane] to avoid LDS bank conflicts.
// ---------------------------------------------------------------------------
__device__ __forceinline__ float chain_lookup(const int4* nodes, int lane,
                                              int head, int coord) {
    float acc = 0.0f, scale = 1.0f;
    int idx = head;
    while (idx >= 0 && scale > EPS_SCALE) {
        int4 nd = nodes[idx * 32 + lane];          // ds_load_b128
        if (nd.y == coord) acc += scale * __int_as_float(nd.z);
        scale *= __int_as_float(nd.w);
        idx = nd.x;
    }
    return acc;
}

__global__ __launch_bounds__(32) void k_build(const int*   __restrict__ event,
                                              const float* __restrict__ ts,
                                              const int*   __restrict__ tt,
                                              const int*   __restrict__ length,
                                              int*         __restrict__ counts,
                                              uint2*       __restrict__ entries) {
    extern __shared__ char smem[];
    int4* nodes = (int4*)smem;                     // 32 lanes * 512 nodes * 16B
    const int lane = threadIdx.x;
    const int seq  = blockIdx.x * 32 + lane;
    if (seq >= SN) return;

    int head[RR];
#pragma unroll
    for (int r = 0; r < RR; ++r) head[r] = -1;

    // Initial delta: T2G[seq, 0, p0, x0, y0] += PFRAC*(GMAX-GMIN)
    {
        int b  = (seq * SPK + 0) * 3;
        int x0 = event[b + 0], y0 = event[b + 1], p0 = event[b + 2];
        int c0 = (p0 << 14) + (x0 << 7) + y0;
        nodes[0 * 32 + lane] =
            make_int4(-1, c0, __float_as_int(PFRAC * (GMAXV - GMINV)),
                      __float_as_int(1.0f));
        head[0] = 0;
    }

    const int len = length[seq];
    float ts_prev = ts[seq * SPK + 0];
    int   tt_prev = tt[seq * SPK + 0];             // == 0 per setup

    for (int n = 1; n < SPK; ++n) {
        int b  = (seq * SPK + n) * 3;
        int ex = event[b + 0], ey = event[b + 1], ep = event[b + 2];
        float ts_cur = ts[seq * SPK + n];
        int   tt_cur = tt[seq * SPK + n];
        float m = __expf((ts_prev - ts_cur) * TAU_INV);   // decay < 1
        int coord = (ep << 14) + (ex << 7) + ey;

        int   hp  = head[tt_prev];
        float u   = chain_lookup(nodes, lane, hp, coord); // value above GMIN
        float old = m * u + GMINV;
        float add = (len > n) ? PFRAC * (GMAXV - old) : 0.0f;

        nodes[n * 32 + lane] =
            make_int4(hp, coord, __float_as_int(add), __float_as_int(m));
        head[tt_cur] = n;
        ts_prev = ts_cur;
        tt_prev = tt_cur;
    }

    // Flatten each slot's chain into an effective (coord, weight) list.
    for (int r = 0; r < RR; ++r) {
        int sl = seq * RR + r;
        uint2* eb = entries + (size_t)sl * SPK;
        int cnt = 0;
        float scale = 1.0f;
        int idx = head[r];
        while (idx >= 0 && scale > EPS_SCALE) {
            int4 nd = nodes[idx * 32 + lane];
            float w = scale * __int_as_float(nd.z);
            if (w > EPS_SCALE) {
                eb[cnt] = make_uint2((unsigned)nd.y, __float_as_uint(w));
                ++cnt;
            }
            scale *= __int_as_float(nd.w);
            idx = nd.x;
        }
        counts[sl] = cnt;
    }
}

// ---------------------------------------------------------------------------
// Kernel 2: one block per (seq, slot). TDM-load the entry list into LDS while
// zeroing a 128KB dense accumulator, ds_add_f32 scatter, then 2x2 max-pool.
// ---------------------------------------------------------------------------
__global__ __launch_bounds__(256) void k_pool(const int*   __restrict__ counts,
                                              const uint2* __restrict__ entries,
                                              float*       __restrict__ out) {
    extern __shared__ char smem[];
    float* dense = (float*)smem;                        // 32768 floats (128KB)
    uint2* ebuf  = (uint2*)(smem + NCOORD * sizeof(float)); // 512 entries (4KB)

    const int tid = threadIdx.x;
    const int bid = blockIdx.x;                         // seq*RR + r
    int cnt = __builtin_amdgcn_readfirstlane(counts[bid]);

    // Async Tensor Data Mover: copy 2*cnt dwords of entry data into LDS.
    // D# per CDNA5 ISA ch.8: group0 = {flags, lds_addr, global_addr, type=2},
    // group1 = {data_size=4B, tensor_dim0=n32 x dim1=1, tile 1 x n32}.
    if (tid < 32) {
        const uint2* src = entries + (size_t)bid * SPK;
        unsigned lds_addr = (unsigned)(uintptr_t)ebuf;  // low 32b = LDS offset
        uint64_t ga  = (uint64_t)(uintptr_t)src;
        unsigned n32 = (unsigned)cnt * 2u;              // dwords (tile_dim0=0 -> NOP)
        v4u g0;
        g0.x = 1u;                                      // count=1 valid descriptor
        g0.y = lds_addr;
        g0.z = (unsigned)ga;                            // global_addr[31:0]
        g0.w = (unsigned)(ga >> 32) | (2u << 30);       // global_addr[56:32], type=2
        v8i g1;
        g1[0] = (int)(2u << 16);                        // data_size = 4 bytes
        g1[1] = (int)((n32 & 0xFFFFu) << 16);           // tensor_dim0[15:0]
        g1[2] = (int)((n32 >> 16) | (1u << 16));        // tensor_dim0[31:16], dim1=1
        g1[3] = (int)((n32 & 0xFFFFu) << 16);           // tile_dim0 = n32
        g1[4] = 1;                                      // tile_dim1 = 1
        g1[5] = (int)n32;                               // tensor_dim0_stride
        g1[6] = 0;
        g1[7] = 0;
        v4i gz4 = {0, 0, 0, 0};                         // groups 2/3 unused (2D)
        v8i gz8 = {0, 0, 0, 0, 0, 0, 0, 0};
        __builtin_amdgcn_tensor_load_to_lds(g0, g1, gz4, gz4, gz8, 0);
    }

    // Zero the dense accumulator while the TDM copy is in flight.
    float4* dv = (float4*)dense;
#pragma unroll
    for (int i = 0; i < 32; ++i)
        dv[tid + i * 256] = make_float4(0.f, 0.f, 0.f, 0.f);

    if (tid < 32) __builtin_amdgcn_s_wait_tensorcnt(0);
    __syncthreads();

    // Scatter-accumulate (duplicate coords possible -> LDS float atomics).
    for (int i = tid; i < cnt; i += 256) {
        uint2 e = ebuf[i];
        atomicAdd(&dense[e.x], __uint_as_float(e.y));
    }
    __syncthreads();

    // 2x2 max-pool: T2G = GMIN + dense (dense >= 0 everywhere).
    float* ob = out + (size_t)bid * (PP * 64 * 64);
    for (int idx = tid; idx < PP * 64 * 64; idx += 256) {
        int p   = idx >> 12;
        int rem = idx & 4095;
        int i   = rem >> 6, j = rem & 63;
        const float* dp = dense + p * (HH * WW) + (i * 2) * WW + j * 2;
        float a = dp[0], b = dp[1], c = dp[WW], d = dp[WW + 1];
        ob[idx] = GMINV + fmaxf(fmaxf(a, b), fmaxf(c, d));
    }
}

// ---------------------------------------------------------------------------
extern "C" void kernel_launch(void* const* d_in, const int* in_sizes, int n_in,
                              void* d_out, int out_size, void* d_ws, size_t ws_size,
                              hipStream_t stream) {
    const int*   event  = (const int*)d_in[0];   // (SN, SPK, 3) int32
    const float* ts     = (const float*)d_in[1]; // (SN, SPK) f32
    const int*   tt     = (const int*)d_in[2];   // (SN, SPK) int32
    const int*   length = (const int*)d_in[3];   // (SN,) int32
    float* out = (float*)d_out;                  // (SN, RR, P, 64, 64) f32

    char*  ws      = (char*)d_ws;
    int*   counts  = (int*)ws;                   // SN*RR ints      (4 KB)
    uint2* entries = (uint2*)(ws + 4096);        // SN*RR*SPK uint2 (4 MB)

    // Phase 1: chains in 256KB dynamic LDS (32 lanes x 512 nodes x 16B).
    k_build<<<dim3(SN / 32), dim3(32), 32 * SPK * 16, stream>>>(
        event, ts, tt, length, counts, entries);

    // Phase 2: 128KB dense accumulator + 4KB entry staging per block.
    k_pool<<<dim3(SN * RR), dim3(256),
             NCOORD * sizeof(float) + SPK * sizeof(uint2), stream>>>(
        counts, entries, out);
}